// VGAE_19439021982274
// MI455X (gfx1250) — compile-verified
//
#include <hip/hip_runtime.h>
#include <math.h>

#define NN 8192
#define EE 131072
#define DD 256
#define HH 32
#define LL 16
#define CC 7
#define COVC 0.1f
#define LN2PIF 1.8378770664093453f

typedef float v2f __attribute__((ext_vector_type(2)));
typedef float v8f __attribute__((ext_vector_type(8)));

__device__ __forceinline__ float wave_reduce_sum(float v) {
    // wave32 butterfly reduction
    #pragma unroll
    for (int m = 16; m >= 1; m >>= 1) v += __shfl_xor(v, m, 32);
    return v;
}

// ---------------------------------------------------------------------------
// K1: degrees + adjacency bitmap (handles duplicate edges like .set(1.0))
// ---------------------------------------------------------------------------
__global__ void k_edges(const int* __restrict__ src, const int* __restrict__ dst,
                        float* __restrict__ degO, float* __restrict__ degI,
                        unsigned* __restrict__ bitmap) {
    int e = blockIdx.x * blockDim.x + threadIdx.x;
    if (e >= EE) return;
    int s = src[e], d = dst[e];
    atomicAdd(&degO[s], 1.0f);
    atomicAdd(&degI[d], 1.0f);
    unsigned idx = (unsigned)s * NN + (unsigned)d;   // < 2^26
    atomicOr(&bitmap[idx >> 5], 1u << (idx & 31));
}

// ---------------------------------------------------------------------------
// K2: norms = rsqrt(max(deg,1)) in place
// ---------------------------------------------------------------------------
__global__ void k_norms(float* __restrict__ degO, float* __restrict__ degI) {
    int n = blockIdx.x * blockDim.x + threadIdx.x;
    if (n >= NN) return;
    degO[n] = rsqrtf(fmaxf(degO[n], 1.0f));
    degI[n] = rsqrtf(fmaxf(degI[n], 1.0f));
}

// ---------------------------------------------------------------------------
// K3: m1 = (x * out_norm) @ W1   [8192,256]@[256,32] via V_WMMA_F32_16X16X4_F32
// One wave per 16x16 output tile; 1024 tiles = 128 blocks x 8 waves exactly.
// fp32 A 16x4 layout: vgpr v, half hi -> K = k0 + v + 2*hi, M/N = lane&15.
// ---------------------------------------------------------------------------
__global__ void k_gemm1(const float* __restrict__ x, const float* __restrict__ W1,
                        const float* __restrict__ outn, float* __restrict__ m1) {
    int wave = threadIdx.x >> 5;
    int lane = threadIdx.x & 31;
    int hi   = lane >> 4;
    int ln   = lane & 15;
    int tile = blockIdx.x * 8 + wave;      // 0..1023, exact coverage
    int it = tile >> 1;                    // M tile (0..511)
    int jt = tile & 1;                     // N tile (0..1): H=32

    int row = it * 16 + ln;
    int col = jt * 16 + ln;
    float scale = outn[row];

    v8f c = {};
    #pragma unroll 4
    for (int k0 = 0; k0 < DD; k0 += 4) {
        v2f a, b;
        int ka = k0 + 2 * hi;
        a.x = x[row * DD + ka]     * scale;
        a.y = x[row * DD + ka + 1] * scale;
        b.x = W1[(ka)     * HH + col];
        b.y = W1[(ka + 1) * HH + col];
        c = __builtin_amdgcn_wmma_f32_16x16x4_f32(false, a, false, b,
                                                  (short)0, c, false, false);
    }
    #pragma unroll
    for (int v = 0; v < 8; ++v) {
        int m = it * 16 + v + 8 * hi;
        m1[m * HH + jt * 16 + ln] = c[v];
    }
}

// ---------------------------------------------------------------------------
// K4: h1acc[dst] += m1[src]  (segment_sum over edges, 32 feats)
// ---------------------------------------------------------------------------
__global__ void k_agg_h(const int* __restrict__ src, const int* __restrict__ dst,
                        const float* __restrict__ m1, float* __restrict__ h1) {
    long long t = (long long)blockIdx.x * blockDim.x + threadIdx.x;
    int e = (int)(t >> 5);
    int f = (int)(t & 31);
    atomicAdd(&h1[dst[e] * HH + f], m1[src[e] * HH + f]);
}

// ---------------------------------------------------------------------------
// K5: mu_m / ls_m = (relu(h1acc*in_norm) * out_norm) @ {Wmu, Wsigma}  (K=32, tiny)
// ---------------------------------------------------------------------------
__global__ void k_gemm2(const float* __restrict__ h1, const float* __restrict__ inn,
                        const float* __restrict__ outn, const float* __restrict__ Wmu,
                        const float* __restrict__ Wsg, float* __restrict__ mum,
                        float* __restrict__ lsm) {
    int t = blockIdx.x * blockDim.x + threadIdx.x;   // NN*LL threads
    int n = t >> 4;
    int l = t & 15;
    float inv = inn[n], ov = outn[n];
    float smu = 0.f, ssg = 0.f;
    #pragma unroll
    for (int k = 0; k < HH; ++k) {
        float hv = fmaxf(h1[n * HH + k] * inv, 0.0f) * ov;
        smu += hv * Wmu[k * LL + l];
        ssg += hv * Wsg[k * LL + l];
    }
    mum[n * LL + l] = smu;
    lsm[n * LL + l] = ssg;
}

// ---------------------------------------------------------------------------
// K6: aggregate mu/log_sigma messages over edges
// ---------------------------------------------------------------------------
__global__ void k_agg_ml(const int* __restrict__ src, const int* __restrict__ dst,
                         const float* __restrict__ mum, const float* __restrict__ lsm,
                         float* __restrict__ mua, float* __restrict__ lsa) {
    long long t = (long long)blockIdx.x * blockDim.x + threadIdx.x;
    int e = (int)(t >> 4);
    int l = (int)(t & 15);
    int di = dst[e] * LL + l, si = src[e] * LL + l;
    atomicAdd(&mua[di], mum[si]);
    atomicAdd(&lsa[di], lsm[si]);
}

// ---------------------------------------------------------------------------
// K7: per node: z = mu + exp(ls)*eps ; S1 += log_p_z - log_p_zax
// ---------------------------------------------------------------------------
__global__ void k_node(const float* __restrict__ mua, const float* __restrict__ lsa,
                       const float* __restrict__ inn, const float* __restrict__ eps,
                       const float* __restrict__ pmu, float* __restrict__ z,
                       float* __restrict__ sc) {
    int n = blockIdx.x * blockDim.x + threadIdx.x;
    if (n >= NN) return;
    float inv = inn[n];
    float zr[LL];
    float lq = 0.f;
    #pragma unroll
    for (int l = 0; l < LL; ++l) {
        float mv = mua[n * LL + l] * inv;
        float lsv = lsa[n * LL + l] * inv;
        float ev = eps[n * LL + l];
        float zv = mv + expf(lsv) * ev;
        zr[l] = zv;
        z[n * LL + l] = zv;
        lq += -lsv - 0.5f * LN2PIF - 0.5f * ev * ev;
    }
    // log p(z): mixture of CC isotropic gaussians, cov = COVC*I, uniform weights
    const float kconst = (float)LL * (LN2PIF + logf(COVC));
    float comp[CC];
    float mx = -3.4e38f;
    #pragma unroll
    for (int c = 0; c < CC; ++c) {
        float d2 = 0.f;
        #pragma unroll
        for (int l = 0; l < LL; ++l) {
            float d = zr[l] - pmu[c * LL + l];
            d2 += d * d;
        }
        comp[c] = -0.5f * (kconst + d2 / COVC);
        mx = fmaxf(mx, comp[c]);
    }
    float se = 0.f;
    #pragma unroll
    for (int c = 0; c < CC; ++c) se += expf(comp[c] - mx);
    float lpz = mx + logf(se) - logf((float)CC);
    atomicAdd(&sc[0], lpz - lq);
}

// ---------------------------------------------------------------------------
// K8: fused z@z^T (fp32 WMMA) + BCE-with-logits sum.
// Grid: 2048 blocks x 8 waves.
//   blockIdx>>5  in [0,64): row-tile group; wave w -> row tile it = group*8+w
//   blockIdx&31  in [0,32): column chunk of 256 columns (16 tiles)
// Block stages the 256x16 column slab of z into LDS (16 KB) once via
// GLOBAL_LOAD_ASYNC_TO_LDS_B128 (ASYNCcnt), then each wave keeps its A
// fragments in registers and sweeps 16 column tiles reading B from LDS.
// ---------------------------------------------------------------------------
__global__ void k_bce(const float* __restrict__ z, const unsigned* __restrict__ bitmap,
                      float* __restrict__ sc) {
    __shared__ float sB[256 * LL];     // 16 KB column slab: rows cb*256..+255
    __shared__ float red[8];
    int wave = threadIdx.x >> 5;
    int lane = threadIdx.x & 31;
    int hi   = lane >> 4;
    int ln   = lane & 15;
    int cb   = blockIdx.x & 31;        // column chunk
    int ig   = blockIdx.x >> 5;        // row group
    int it   = ig * 8 + wave;          // row tile (0..511)

    // ---- async stage: thread t copies z row (cb*256 + t) -> sB[t*16 ..] ----
    {
        int t = threadIdx.x;
        const float* gsrc = z + (size_t)(cb * 256 + t) * LL;
        unsigned ldst = (unsigned)(uintptr_t)(&sB[t * LL]);   // LDS offset = addr[31:0]
        unsigned long long ga = (unsigned long long)(uintptr_t)gsrc;
        asm volatile("global_load_async_to_lds_b128 %0, %1, off\n\t"
                     "global_load_async_to_lds_b128 %0, %1, off offset:16\n\t"
                     "global_load_async_to_lds_b128 %0, %1, off offset:32\n\t"
                     "global_load_async_to_lds_b128 %0, %1, off offset:48"
                     :: "v"(ldst), "v"(ga) : "memory");
        asm volatile("s_wait_asynccnt 0x0" ::: "memory");
    }
    __syncthreads();

    // ---- A fragments for this wave's row tile, kept in registers ----
    int rowA = it * 16 + ln;
    v2f a0, a1, a2, a3;
    {
        const float* zr = z + (size_t)rowA * LL;
        a0.x = zr[0  + 2 * hi]; a0.y = zr[1  + 2 * hi];
        a1.x = zr[4  + 2 * hi]; a1.y = zr[5  + 2 * hi];
        a2.x = zr[8  + 2 * hi]; a2.y = zr[9  + 2 * hi];
        a3.x = zr[12 + 2 * hi]; a3.y = zr[13 + 2 * hi];
    }

    float bsum = 0.f;
    #pragma unroll 1
    for (int i = 0; i < 16; ++i) {     // 16 column tiles in this chunk
        const float* br = &sB[(i * 16 + ln) * LL];
        v8f c = {};
        v2f b;
        b.x = br[0  + 2 * hi]; b.y = br[1  + 2 * hi];
        c = __builtin_amdgcn_wmma_f32_16x16x4_f32(false, a0, false, b, (short)0, c, false, false);
        b.x = br[4  + 2 * hi]; b.y = br[5  + 2 * hi];
        c = __builtin_amdgcn_wmma_f32_16x16x4_f32(false, a1, false, b, (short)0, c, false, false);
        b.x = br[8  + 2 * hi]; b.y = br[9  + 2 * hi];
        c = __builtin_amdgcn_wmma_f32_16x16x4_f32(false, a2, false, b, (short)0, c, false, false);
        b.x = br[12 + 2 * hi]; b.y = br[13 + 2 * hi];
        c = __builtin_amdgcn_wmma_f32_16x16x4_f32(false, a3, false, b, (short)0, c, false, false);

        int col = cb * 256 + i * 16 + ln;
        #pragma unroll
        for (int v = 0; v < 8; ++v) {
            int row = it * 16 + v + 8 * hi;
            float l = c[v];
            unsigned idx = (unsigned)row * NN + (unsigned)col;
            float a = (float)((bitmap[idx >> 5] >> (idx & 31)) & 1u);
            // max(l,0) - l*a + log1p(exp(-|l|))
            bsum += fmaxf(l, 0.f) - l * a + log1pf(expf(-fabsf(l)));
        }
    }
    float wsum = wave_reduce_sum(bsum);
    if (lane == 0) red[wave] = wsum;
    __syncthreads();
    if (threadIdx.x == 0) {
        float tot = 0.f;
        #pragma unroll
        for (int w = 0; w < 8; ++w) tot += red[w];
        atomicAdd(&sc[1], tot);
    }
}

// ---------------------------------------------------------------------------
// K9: out = -(S1 - S2)/N
// ---------------------------------------------------------------------------
__global__ void k_final(const float* __restrict__ sc, float* __restrict__ out) {
    out[0] = -(sc[0] - sc[1]) * (1.0f / (float)NN);
}

extern "C" void kernel_launch(void* const* d_in, const int* in_sizes, int n_in,
                              void* d_out, int out_size, void* d_ws, size_t ws_size,
                              hipStream_t stream) {
    const float* x   = (const float*)d_in[0];
    const float* W1  = (const float*)d_in[1];
    const float* Wmu = (const float*)d_in[2];
    const float* Wsg = (const float*)d_in[3];
    const float* pmu = (const float*)d_in[4];
    const float* eps = (const float*)d_in[5];
    const int*   src = (const int*)d_in[6];
    const int*   dst = (const int*)d_in[7];
    float* out = (float*)d_out;

    char* base = (char*)d_ws;
    size_t off = 0;
    float* degO = (float*)(base + off); off += (size_t)NN * 4;
    float* degI = (float*)(base + off); off += (size_t)NN * 4;
    float* sc   = (float*)(base + off); off += 256;
    float* m1   = (float*)(base + off); off += (size_t)NN * HH * 4;
    float* h1   = (float*)(base + off); off += (size_t)NN * HH * 4;
    float* mum  = (float*)(base + off); off += (size_t)NN * LL * 4;
    float* lsm  = (float*)(base + off); off += (size_t)NN * LL * 4;
    float* mua  = (float*)(base + off); off += (size_t)NN * LL * 4;
    float* lsa  = (float*)(base + off); off += (size_t)NN * LL * 4;
    float* zbuf = (float*)(base + off); off += (size_t)NN * LL * 4;
    unsigned* bitmap = (unsigned*)(base + off); off += (size_t)NN * NN / 8;

    hipMemsetAsync(degO, 0, (size_t)NN * 4, stream);
    hipMemsetAsync(degI, 0, (size_t)NN * 4, stream);
    hipMemsetAsync(sc,   0, 256, stream);
    hipMemsetAsync(h1,   0, (size_t)NN * HH * 4, stream);
    hipMemsetAsync(mua,  0, (size_t)NN * LL * 4, stream);
    hipMemsetAsync(lsa,  0, (size_t)NN * LL * 4, stream);
    hipMemsetAsync(bitmap, 0, (size_t)NN * NN / 8, stream);

    k_edges <<<EE / 256, 256, 0, stream>>>(src, dst, degO, degI, bitmap);
    k_norms <<<NN / 256, 256, 0, stream>>>(degO, degI);
    k_gemm1 <<<128, 256, 0, stream>>>(x, W1, degO, m1);
    k_agg_h <<<(EE * HH) / 256, 256, 0, stream>>>(src, dst, m1, h1);
    k_gemm2 <<<(NN * LL) / 256, 256, 0, stream>>>(h1, degI, degO, Wmu, Wsg, mum, lsm);
    k_agg_ml<<<(EE * LL) / 256, 256, 0, stream>>>(src, dst, mum, lsm, mua, lsa);
    k_node  <<<NN / 256, 256, 0, stream>>>(mua, lsa, degI, eps, pmu, zbuf, sc);
    k_bce   <<<2048, 256, 0, stream>>>(zbuf, bitmap, sc);
    k_final <<<1, 1, 0, stream>>>(sc, out);
}